// MyGRU_GCN_Model_motion_18253611008143
// MI455X (gfx1250) — compile-verified
//
#include <hip/hip_runtime.h>
#include <hip/hip_bf16.h>

#define BATCH 16
#define NVERT 8192
#define NEDGE 131072
#define HDIM  512
#define H2DIM 65536
#define GOUT  8

typedef __attribute__((ext_vector_type(2))) float v2f;
typedef __attribute__((ext_vector_type(4))) float v4f;
typedef __attribute__((ext_vector_type(8))) float v8f;

// D = A x B + C ; A 16x4 f32 (2 VGPR), B 4x16 (2 VGPR), C/D 16x16 f32 (8 VGPR)
__device__ __forceinline__ v8f wmma4(v2f a, v2f b, v8f c) {
    return __builtin_amdgcn_wmma_f32_16x16x4_f32(false, a, false, b,
                                                 (short)0, c, false, false);
}

// Accumulate an 8-wide K chunk with one b128 load per operand and two WMMAs.
// Slot mapping: physical slot (2*half+j) <- actual k = k0 + 4*half + j.
// A and B use the same mapping, so the dot product is exact.
template<int K>
__device__ __forceinline__ v8f acc_k8(const float* __restrict__ xr,
                                      const float* __restrict__ wr,
                                      int k0, int half, v8f acc) {
    v4f a4 = *(const v4f*)(xr + k0 + 4 * half);
    v4f b4 = *(const v4f*)(wr + k0 + 4 * half);
    v2f a0, b0, a1, b1;
    a0.x = a4.x; a0.y = a4.y; a1.x = a4.z; a1.y = a4.w;
    b0.x = b4.x; b0.y = b4.y; b1.x = b4.z; b1.y = b4.w;
    acc = wmma4(a0, b0, acc);
    acc = wmma4(a1, b1, acc);
    return acc;
}

// Branchless tail-safe 4-wide K chunk (clamped index + value mask; no EXEC div.)
template<int K>
__device__ __forceinline__ v8f acc_k4_guard(const float* __restrict__ xr,
                                            const float* __restrict__ wr,
                                            int k0, int half, v8f acc) {
    int kA = k0 + 2 * half;
    int i0 = (kA     < K) ? kA     : (K - 1);
    int i1 = (kA + 1 < K) ? kA + 1 : (K - 1);
    float m0 = (kA     < K) ? 1.f : 0.f;
    float m1 = (kA + 1 < K) ? 1.f : 0.f;
    v2f a, b;
    a.x = xr[i0] * m0;  a.y = xr[i1] * m1;
    b.x = wr[i0] * m0;  b.y = wr[i1] * m1;
    return wmma4(a, b, acc);
}

// ---------------------------------------------------------------------------
// Y[16, Nout] = X[16,K] @ W[Nout,K]^T + bias   (one wave per 16-col tile)
// ---------------------------------------------------------------------------
template<int K>
__global__ void k_gemm16(const float* __restrict__ X,
                         const float* __restrict__ W,
                         const float* __restrict__ bias,
                         float* __restrict__ Y,
                         int Nout)
{
    int wave = (int)(blockIdx.x * blockDim.x + threadIdx.x) >> 5;
    int lane = threadIdx.x & 31;
    int n0   = wave * 16;
    if (n0 >= Nout) return;
    int half = lane >> 4;
    int idx  = lane & 15;

    const float* xr = X + (size_t)idx * K;
    const float* wr = W + (size_t)(n0 + idx) * K;

    v8f acc = {};
    if constexpr (K % 8 == 0) {
#pragma unroll 8
        for (int k0 = 0; k0 < K; k0 += 8)
            acc = acc_k8<K>(xr, wr, k0, half, acc);
    } else {
#pragma unroll
        for (int k0 = 0; k0 < K; k0 += 4)
            acc = acc_k4_guard<K>(xr, wr, k0, half, acc);
    }
    float bn = bias[n0 + idx];
#pragma unroll
    for (int r = 0; r < 8; ++r) {
        int m = r + 8 * half;
        Y[(size_t)m * Nout + n0 + idx] = acc[r] + bn;
    }
}

// ---------------------------------------------------------------------------
// Y[R,64] = X[R,K] @ W[64,K]^T   (no bias) — GCN feature transform.
// ---------------------------------------------------------------------------
template<int K>
__global__ void k_gemm_rows(const float* __restrict__ X,
                            const float* __restrict__ W,
                            float* __restrict__ Y,
                            int R)
{
    int wave = (int)(blockIdx.x * blockDim.x + threadIdx.x) >> 5;
    int lane = threadIdx.x & 31;
    int rt = wave >> 2;
    int ct = wave & 3;
    if (rt * 16 >= R) return;
    int half = lane >> 4;
    int idx  = lane & 15;

    const float* xr = X + (size_t)(rt * 16 + idx) * K;
    const float* wr = W + (size_t)(ct * 16 + idx) * K;

    v8f acc = {};
    if constexpr (K % 8 == 0) {
#pragma unroll
        for (int k0 = 0; k0 < K; k0 += 8)
            acc = acc_k8<K>(xr, wr, k0, half, acc);
    } else {
#pragma unroll
        for (int k0 = 0; k0 < K; k0 += 4)
            acc = acc_k4_guard<K>(xr, wr, k0, half, acc);
    }
#pragma unroll
    for (int r = 0; r < 8; ++r) {
        int m = r + 8 * half;
        Y[(size_t)(rt * 16 + m) * 64 + ct * 16 + idx] = acc[r];
    }
}

// ---------------------------------------------------------------------------
// GRU gate fusion: nh = (1-z)*n + z*h
// ---------------------------------------------------------------------------
__global__ void k_gru_gate(const float* __restrict__ gi,
                           const float* __restrict__ gh,
                           const float* __restrict__ h,
                           float* __restrict__ out, int H)
{
    int i = blockIdx.x * blockDim.x + threadIdx.x;
    if (i >= BATCH * H) return;
    int b = i / H, j = i - b * H;
    const float* gib = gi + (size_t)b * 3 * H;
    const float* ghb = gh + (size_t)b * 3 * H;
    float r = 1.f / (1.f + __expf(-(gib[j]         + ghb[j])));
    float z = 1.f / (1.f + __expf(-(gib[H + j]     + ghb[H + j])));
    float n = tanhf(gib[2 * H + j] + r * ghb[2 * H + j]);
    out[i] = (1.f - z) * n + z * h[i];
}

// ---------------------------------------------------------------------------
// PReLU + eval-mode BatchNorm: y = prelu(x)/sqrt(1+eps)*g + be
// ---------------------------------------------------------------------------
__global__ void k_mlp_post(const float* __restrict__ pre,
                           const float* __restrict__ a,
                           const float* __restrict__ g,
                           const float* __restrict__ be,
                           float* __restrict__ out, int Nout)
{
    int i = blockIdx.x * blockDim.x + threadIdx.x;
    if (i >= BATCH * Nout) return;
    int c = i % Nout;
    float v = pre[i];
    v = (v >= 0.f) ? v : a[c] * v;
    const float inv = 0.9999950000374997f;   // 1/sqrt(1+1e-5)
    out[i] = v * inv * g[c] + be[c];
}

// ---------------------------------------------------------------------------
// Degree / norm helpers
// ---------------------------------------------------------------------------
__global__ void k_zero(float* __restrict__ p, int n)
{
    int i = blockIdx.x * blockDim.x + threadIdx.x;
    if (i < n) p[i] = 0.f;
}

__global__ void k_deg(const int* __restrict__ dst, float* __restrict__ deg)
{
    int e = blockIdx.x * blockDim.x + threadIdx.x;
    if (e < NEDGE) atomicAdd(&deg[dst[e]], 1.f);
}

__global__ void k_dinv(float* __restrict__ deg)
{
    int i = blockIdx.x * blockDim.x + threadIdx.x;
    if (i < NVERT) deg[i] = rsqrtf(deg[i] + 1.f);
}

__global__ void k_norm(const int* __restrict__ src, const int* __restrict__ dst,
                       const float* __restrict__ dinv, float* __restrict__ nrm)
{
    int e = blockIdx.x * blockDim.x + threadIdx.x;
    if (e < NEDGE) nrm[e] = dinv[src[e]] * dinv[dst[e]];
}

// x[b,n,f] = hw[b,n,f] * dinv[n]^2 + bias[f]  (self-loop + bias, float4)
__global__ void k_agg_init(const v4f* __restrict__ hw,
                           const float* __restrict__ dinv,
                           const v4f* __restrict__ bias,
                           v4f* __restrict__ x)
{
    int i = blockIdx.x * blockDim.x + threadIdx.x;   // float4 index
    if (i >= BATCH * NVERT * 16) return;
    int q = i & 15;
    int n = (i >> 4) & (NVERT - 1);
    float d = dinv[n];
    v4f h = hw[i], b = bias[q];
    v4f r;
    r.x = h.x * d * d + b.x;
    r.y = h.y * d * d + b.y;
    r.z = h.z * d * d + b.z;
    r.w = h.w * d * d + b.w;
    x[i] = r;
}

// One block per edge; thread t handles float4 q of batch b (256 = 16*16 exact)
__global__ void k_agg_edges(const int* __restrict__ src, const int* __restrict__ dst,
                            const float* __restrict__ nrm,
                            const float* __restrict__ hw, float* __restrict__ x)
{
    int e = blockIdx.x;
    int s = src[e], d = dst[e];
    float w = nrm[e];
    int t = threadIdx.x;                  // 0..255
    int b = t >> 4, q = t & 15;
    const v4f* hv = (const v4f*)(hw + ((size_t)b * NVERT + s) * 64) + q;
    float* xo = x + ((size_t)b * NVERT + d) * 64 + q * 4;
    v4f h = *hv;
    atomicAdd(xo + 0, h.x * w);
    atomicAdd(xo + 1, h.y * w);
    atomicAdd(xo + 2, h.z * w);
    atomicAdd(xo + 3, h.w * w);
}

// ---------------------------------------------------------------------------
// Output projection: y[b, n*3+c] = [gru_out(8) | x1(64)] . Wo[c,:] + bo[c]
// ---------------------------------------------------------------------------
__global__ void k_out(const float* __restrict__ h2, const float* __restrict__ x1,
                      const float* __restrict__ Wo, const float* __restrict__ bo,
                      float* __restrict__ y)
{
    int i = blockIdx.x * blockDim.x + threadIdx.x;   // b*NVERT + n
    if (i >= BATCH * NVERT) return;
    int b = i >> 13;
    int n = i & (NVERT - 1);
    const float* go = h2 + (size_t)b * H2DIM + (size_t)n * GOUT;
    const float* xv = x1 + (size_t)i * 64;
#pragma unroll
    for (int c = 0; c < 3; ++c) {
        float s = bo[c];
        const float* w = Wo + c * 72;
#pragma unroll
        for (int j = 0; j < 8; ++j) s += go[j] * w[j];
#pragma unroll
        for (int j = 0; j < 64; ++j) s += xv[j] * w[8 + j];
        y[(size_t)b * (3 * NVERT) + (size_t)n * 3 + c] = s;
    }
}

static inline int cdiv(int a, int b) { return (a + b - 1) / b; }

extern "C" void kernel_launch(void* const* d_in, const int* in_sizes, int n_in,
                              void* d_out, int out_size, void* d_ws, size_t ws_size,
                              hipStream_t stream)
{
    (void)in_sizes; (void)n_in; (void)out_size; (void)ws_size;

    const float* x      = (const float*)d_in[0];
    const float* mdata  = (const float*)d_in[1];
    const float* verts  = (const float*)d_in[2];
    const float* hidden = (const float*)d_in[3];
    const float* mhid   = (const float*)d_in[4];
    const int*   esrc   = (const int*)d_in[5];
    const int*   edst   = (const int*)d_in[6];
    const float* W_ih = (const float*)d_in[7];   const float* b_ih = (const float*)d_in[8];
    const float* W_hh = (const float*)d_in[9];   const float* b_hh = (const float*)d_in[10];
    const float* W1 = (const float*)d_in[11];    const float* b1 = (const float*)d_in[12];
    const float* a1 = (const float*)d_in[13];    const float* g1 = (const float*)d_in[14];
    const float* be1 = (const float*)d_in[15];
    const float* W2 = (const float*)d_in[16];    const float* b2 = (const float*)d_in[17];
    const float* a2 = (const float*)d_in[18];    const float* g2 = (const float*)d_in[19];
    const float* be2 = (const float*)d_in[20];
    const float* Wo = (const float*)d_in[21];    const float* bo = (const float*)d_in[22];
    const float* Wg1 = (const float*)d_in[23];   const float* bg1 = (const float*)d_in[24];
    const float* Wg2 = (const float*)d_in[25];   const float* bg2 = (const float*)d_in[26];
    const float* Wm_ih = (const float*)d_in[27]; const float* bm_ih = (const float*)d_in[28];
    const float* Wm_hh = (const float*)d_in[29]; const float* bm_hh = (const float*)d_in[30];
    // d_in[31]/[32] (Wml/bml) feed edge_weights, which the reference deletes: skip.

    float* out = (float*)d_out;
    float* y   = out;                       // [16, 3N]  = 393216
    float* nh  = out + 393216;              // [16, 512] =   8192
    float* nmh = out + 401408;              // [16, 128] =   2048

    float* ws = (float*)d_ws;
    size_t off = 0;
    float* gi_m = ws + off; off += (size_t)BATCH * 384;
    float* gh_m = ws + off; off += (size_t)BATCH * 384;
    float* gi   = ws + off; off += (size_t)BATCH * 1536;
    float* gh   = ws + off; off += (size_t)BATCH * 1536;
    float* h1p  = ws + off; off += (size_t)BATCH * 512;
    float* h1   = ws + off; off += (size_t)BATCH * 512;
    float* h2p  = ws + off; off += (size_t)BATCH * H2DIM;
    float* h2   = ws + off; off += (size_t)BATCH * H2DIM;
    float* dinv = ws + off; off += NVERT;
    float* nrm  = ws + off; off += NEDGE;
    float* bufA = ws + off; off += (size_t)BATCH * NVERT * 64;   // hw1 / hw2
    float* bufB = ws + off; off += (size_t)BATCH * NVERT * 64;   // x0  / x1

    const int TB = 256;
    // --- motion GRU (edge_weights branch is dead code in the reference) ---
    k_gemm16<63> <<<cdiv(24 * 32, TB), TB, 0, stream>>>(mdata, Wm_ih, bm_ih, gi_m, 384);
    k_gemm16<128><<<cdiv(24 * 32, TB), TB, 0, stream>>>(mhid,  Wm_hh, bm_hh, gh_m, 384);
    k_gru_gate<<<cdiv(BATCH * 128, TB), TB, 0, stream>>>(gi_m, gh_m, mhid, nmh, 128);

    // --- main GRU ---
    k_gemm16<96> <<<cdiv(96 * 32, TB), TB, 0, stream>>>(x,      W_ih, b_ih, gi, 1536);
    k_gemm16<512><<<cdiv(96 * 32, TB), TB, 0, stream>>>(hidden, W_hh, b_hh, gh, 1536);
    k_gru_gate<<<cdiv(BATCH * HDIM, TB), TB, 0, stream>>>(gi, gh, hidden, nh, HDIM);

    // --- MLP: 512 -> 512 -> 65536 (W2 read = 128 MB: HBM-bound stage) ---
    k_gemm16<512><<<cdiv(32 * 32, TB), TB, 0, stream>>>(nh, W1, b1, h1p, 512);
    k_mlp_post<<<cdiv(BATCH * 512, TB), TB, 0, stream>>>(h1p, a1, g1, be1, h1, 512);
    k_gemm16<512><<<cdiv(4096 * 32, TB), TB, 0, stream>>>(h1, W2, b2, h2p, H2DIM);
    k_mlp_post<<<cdiv(BATCH * H2DIM, TB), TB, 0, stream>>>(h2p, a2, g2, be2, h2, H2DIM);

    // --- GCN normalization coefficients ---
    k_zero<<<cdiv(NVERT, TB), TB, 0, stream>>>(dinv, NVERT);
    k_deg<<<cdiv(NEDGE, TB), TB, 0, stream>>>(edst, dinv);
    k_dinv<<<cdiv(NVERT, TB), TB, 0, stream>>>(dinv);
    k_norm<<<cdiv(NEDGE, TB), TB, 0, stream>>>(esrc, edst, dinv, nrm);

    const int R = BATCH * NVERT;                 // 131072 rows
    const int gemmRowsBlocks = cdiv((R / 16) * 4 * 32, TB);
    const int aggInitBlocks  = cdiv(R * 16, TB); // float4 granularity

    // --- GCN layer 1: [B,N,3] -> [B,N,64] ---
    k_gemm_rows<3> <<<gemmRowsBlocks, TB, 0, stream>>>(verts, Wg1, bufA, R);
    k_agg_init<<<aggInitBlocks, TB, 0, stream>>>((const v4f*)bufA, dinv,
                                                 (const v4f*)bg1, (v4f*)bufB);
    k_agg_edges<<<NEDGE, TB, 0, stream>>>(esrc, edst, nrm, bufA, bufB);

    // --- GCN layer 2: [B,N,64] -> [B,N,64] ---
    k_gemm_rows<64><<<gemmRowsBlocks, TB, 0, stream>>>(bufB, Wg2, bufA, R);
    k_agg_init<<<aggInitBlocks, TB, 0, stream>>>((const v4f*)bufA, dinv,
                                                 (const v4f*)bg2, (v4f*)bufB);
    k_agg_edges<<<NEDGE, TB, 0, stream>>>(esrc, edst, nrm, bufA, bufB);

    // --- output projection ---
    k_out<<<cdiv(R, TB), TB, 0, stream>>>(h2, bufB, Wo, bo, y);
}